// GCN_8693013807615
// MI455X (gfx1250) — compile-verified
//
#include <hip/hip_runtime.h>

typedef float v2f __attribute__((ext_vector_type(2)));
typedef float v8f __attribute__((ext_vector_type(8)));

#define HD 64    // hidden dim per layer
#define CD 192   // concat dim = 3*64
#define GN 256   // number of graphs

// ---------------- degree / dinv / counts ----------------
__global__ void k_deg(const int* __restrict__ dst, int E, float* __restrict__ deg) {
  int e = blockIdx.x * blockDim.x + threadIdx.x;
  if (e < E) atomicAdd(&deg[dst[e]], 1.0f);
}

__global__ void k_dinv_cnt(float* __restrict__ dinv, const int* __restrict__ batch,
                           float* __restrict__ cnt, int N) {
  int i = blockIdx.x * blockDim.x + threadIdx.x;
  if (i < N) {
    dinv[i] = rsqrtf(dinv[i] + 1.0f);   // +1 for self-loop; deg >= 1 always
    atomicAdd(&cnt[batch[i]], 1.0f);
  }
}

// ---------------- fp32 WMMA GEMM:  Hout(N x 64) = Z(N x K, stride ldz) * W(K x 64) ----------------
__global__ void k_gemm_wmma(const float* __restrict__ Z, int ldz,
                            const float* __restrict__ W, int K,
                            float* __restrict__ Hout, int N) {
  int wid = blockIdx.x * (blockDim.x >> 5) + (threadIdx.x >> 5);
  int tilesM = (N + 15) >> 4;
  if (wid >= tilesM * 4) return;          // wave-uniform: EXEC stays all-ones below
  int tm = wid >> 2, tn = wid & 3;
  int lane = threadIdx.x & 31;
  int half = lane >> 4;                   // 0: K pair {0,1}; 1: K pair {2,3}
  int m16 = lane & 15;
  int row = tm * 16 + m16;
  int lrow = row < N ? row : N - 1;       // clamp (no branch -> EXEC untouched)
  const float* zr = Z + (size_t)lrow * ldz;
  int col = tn * 16 + m16;
  const float* wp = W + col;
  v8f c = {};
  for (int k0 = 0; k0 < K; k0 += 4) {
    int ka = k0 + 2 * half;
    v2f a, b;
    a.x = zr[ka];
    a.y = zr[ka + 1];
    b.x = wp[(size_t)ka * HD];
    b.y = wp[(size_t)(ka + 1) * HD];
    c = __builtin_amdgcn_wmma_f32_16x16x4_f32(false, a, false, b, (short)0, c, false, false);
  }
  int rbase = tm * 16 + half * 8;
#pragma unroll
  for (int j = 0; j < 8; ++j) {
    int r = rbase + j;
    if (r < N) Hout[(size_t)r * HD + col] = c[j];
  }
}

// ---------------- self-loop init:  agg[i,:] = h[i,:] * dinv[i]^2 ----------------
__global__ void k_selfinit(const float* __restrict__ h, const float* __restrict__ dinv,
                           float* __restrict__ agg, int N) {
  int t = blockIdx.x * blockDim.x + threadIdx.x;
  if (t < N * HD) {
    float di = dinv[t >> 6];
    agg[t] = h[t] * di * di;
  }
}

// ---------------- edge scatter:  agg[dst,:] += h[src,:] * dinv[src]*dinv[dst] ----------------
__global__ void k_edge_scatter(const int* __restrict__ src, const int* __restrict__ dst,
                               const float* __restrict__ dinv, const float* __restrict__ h,
                               float* __restrict__ agg, int E) {
  int t = blockIdx.x * blockDim.x + threadIdx.x;
  if (t >= E * 16) return;
  int e = t >> 4;
  int c4 = (t & 15) << 2;
  int s = src[e], d = dst[e];
  float w = dinv[s] * dinv[d];
  const float4 hv = *(const float4*)(h + (size_t)s * HD + c4);
  float* ap = agg + (size_t)d * HD + c4;
  atomicAdd(ap + 0, hv.x * w);
  atomicAdd(ap + 1, hv.y * w);
  atomicAdd(ap + 2, hv.z * w);
  atomicAdd(ap + 3, hv.w * w);
}

// ---------------- z = relu(agg + b), in place; accumulate BN sums (LDS-reduced) ----------------
__global__ void k_relu_stats(float* __restrict__ agg, const float* __restrict__ bias,
                             float* __restrict__ bn_sum, float* __restrict__ bn_sq, int N) {
  __shared__ float ssum[256], ssq[256];
  int c = threadIdx.x & 63;
  int quad = threadIdx.x >> 6;
  int base = blockIdx.x * 64;
  float b = bias[c];
  float s = 0.f, q = 0.f;
  for (int r = quad; r < 64; r += 4) {
    int row = base + r;
    if (row < N) {
      float v = fmaxf(agg[(size_t)row * HD + c] + b, 0.f);
      agg[(size_t)row * HD + c] = v;
      s += v; q += v * v;
    }
  }
  ssum[threadIdx.x] = s; ssq[threadIdx.x] = q;
  __syncthreads();
  if (quad == 0) {
    float ts = ssum[c] + ssum[64 + c] + ssum[128 + c] + ssum[192 + c];
    float tq = ssq[c] + ssq[64 + c] + ssq[128 + c] + ssq[192 + c];
    atomicAdd(&bn_sum[c], ts);
    atomicAdd(&bn_sq[c], tq);
  }
}

__global__ void k_bnfinal(const float* __restrict__ bn_sum, const float* __restrict__ bn_sq,
                          const float* __restrict__ gamma, const float* __restrict__ beta,
                          float* __restrict__ scale, float* __restrict__ shift, float invN) {
  int c = threadIdx.x;
  float mu = bn_sum[c] * invN;
  float var = fmaxf(bn_sq[c] * invN - mu * mu, 0.f);   // biased var
  float sc = rsqrtf(var + 1e-5f) * gamma[c];
  scale[c] = sc;
  shift[c] = beta[c] - mu * sc;
}

// ---------------- BN apply, write zc column block, segment-pool into gsum ----------------
__global__ void k_bnapply_pool(const float* __restrict__ agg, const float* __restrict__ scale,
                               const float* __restrict__ shift, const int* __restrict__ batch,
                               float* __restrict__ zc, float* __restrict__ gsum,
                               int N, int lofs) {
  int c = threadIdx.x & 63;
  int quad = threadIdx.x >> 6;
  int base = blockIdx.x * 64;
  float sc = scale[c], sh = shift[c];
  int gcur = -1;
  float acc = 0.f;
  for (int r = quad; r < 64; r += 4) {
    int row = base + r;
    if (row < N) {
      float v = agg[(size_t)row * HD + c] * sc + sh;
      zc[(size_t)row * CD + lofs + c] = v;
      int g = batch[row];                       // batch is sorted -> long runs
      if (g != gcur) {
        if (gcur >= 0) atomicAdd(&gsum[(size_t)gcur * CD + lofs + c], acc);
        gcur = g; acc = 0.f;
      }
      acc += v;
    }
  }
  if (gcur >= 0) atomicAdd(&gsum[(size_t)gcur * CD + lofs + c], acc);
}

// ---------------- row L2 normalize (192 wide): one wave per row ----------------
__global__ void k_l2_zc(const float* __restrict__ zc, float* __restrict__ out, int rows) {
  int wid = blockIdx.x * (blockDim.x >> 5) + (threadIdx.x >> 5);
  if (wid >= rows) return;
  int lane = threadIdx.x & 31;
  const float* a = zc + (size_t)wid * CD;
  float v[6]; float s = 0.f;
#pragma unroll
  for (int j = 0; j < 6; ++j) { v[j] = a[lane + 32 * j]; s += v[j] * v[j]; }
#pragma unroll
  for (int off = 16; off >= 1; off >>= 1) s += __shfl_xor(s, off, 32);
  float inv = 1.0f / fmaxf(sqrtf(s), 1e-12f);
  float* o = out + (size_t)wid * CD;
#pragma unroll
  for (int j = 0; j < 6; ++j) o[lane + 32 * j] = v[j] * inv;
}

__global__ void k_l2_gc(const float* __restrict__ gsum, const float* __restrict__ cnt,
                        float* __restrict__ out, int rows) {
  int wid = blockIdx.x * (blockDim.x >> 5) + (threadIdx.x >> 5);
  if (wid >= rows) return;
  int lane = threadIdx.x & 31;
  float icnt = 1.0f / fmaxf(cnt[wid], 1.0f);
  const float* a = gsum + (size_t)wid * CD;
  float v[6]; float s = 0.f;
#pragma unroll
  for (int j = 0; j < 6; ++j) { v[j] = a[lane + 32 * j] * icnt; s += v[j] * v[j]; }
#pragma unroll
  for (int off = 16; off >= 1; off >>= 1) s += __shfl_xor(s, off, 32);
  float inv = 1.0f / fmaxf(sqrtf(s), 1e-12f);
  float* o = out + (size_t)wid * CD;
#pragma unroll
  for (int j = 0; j < 6; ++j) o[lane + 32 * j] = v[j] * inv;
}

extern "C" void kernel_launch(void* const* d_in, const int* in_sizes, int n_in,
                              void* d_out, int out_size, void* d_ws, size_t ws_size,
                              hipStream_t stream) {
  (void)n_in; (void)out_size; (void)ws_size;
  const float* x     = (const float*)d_in[0];
  const int*   ei    = (const int*)d_in[1];
  const int*   batch = (const int*)d_in[2];
  int N = in_sizes[0] / 128;
  int E = in_sizes[1] / 2;
  const int* src = ei;
  const int* dst = ei + E;
  const float* Wl[3]  = {(const float*)d_in[3],  (const float*)d_in[7],  (const float*)d_in[11]};
  const float* bl[3]  = {(const float*)d_in[4],  (const float*)d_in[8],  (const float*)d_in[12]};
  const float* gl[3]  = {(const float*)d_in[5],  (const float*)d_in[9],  (const float*)d_in[13]};
  const float* bel[3] = {(const float*)d_in[6],  (const float*)d_in[10], (const float*)d_in[14]};

  // workspace carve-up (floats)
  float* ws   = (float*)d_ws;
  float* dinv = ws;                       ws += N;                 // degree -> dinv (in place)
  float* h    = ws;                       ws += (size_t)N * HD;    // GEMM output
  float* agg  = ws;                       ws += (size_t)N * HD;    // aggregation / z
  float* zc   = ws;                       ws += (size_t)N * CD;    // concat of BN outputs
  float* gsum = ws;                       ws += (size_t)GN * CD;   // graph pool sums
  float* cnt  = ws;                       ws += GN;                // graph counts
  float* bn   = ws;                       // 256 floats: sum | sq | scale | shift
  float* bn_sum = bn, *bn_sq = bn + 64, *scale = bn + 128, *shift = bn + 192;

  hipMemsetAsync(dinv, 0, (size_t)N * sizeof(float), stream);
  hipMemsetAsync(gsum, 0, (size_t)GN * CD * sizeof(float), stream);
  hipMemsetAsync(cnt,  0, (size_t)GN * sizeof(float), stream);

  k_deg<<<(E + 255) / 256, 256, 0, stream>>>(dst, E, dinv);
  k_dinv_cnt<<<(N + 255) / 256, 256, 0, stream>>>(dinv, batch, cnt, N);

  for (int l = 0; l < 3; ++l) {
    const float* Z = (l == 0) ? x : (zc + (size_t)(l - 1) * HD);
    int ldz = (l == 0) ? 128 : CD;
    int K   = (l == 0) ? 128 : HD;
    int tilesM = (N + 15) / 16;
    int waves = tilesM * 4;
    k_gemm_wmma<<<(waves + 7) / 8, 256, 0, stream>>>(Z, ldz, Wl[l], K, h, N);
    k_selfinit<<<(N * HD + 255) / 256, 256, 0, stream>>>(h, dinv, agg, N);
    k_edge_scatter<<<((size_t)E * 16 + 255) / 256, 256, 0, stream>>>(src, dst, dinv, h, agg, E);
    hipMemsetAsync(bn, 0, 128 * sizeof(float), stream);
    k_relu_stats<<<(N + 63) / 64, 256, 0, stream>>>(agg, bl[l], bn_sum, bn_sq, N);
    k_bnfinal<<<1, 64, 0, stream>>>(bn_sum, bn_sq, gl[l], bel[l], scale, shift, 1.0f / (float)N);
    k_bnapply_pool<<<(N + 63) / 64, 256, 0, stream>>>(agg, scale, shift, batch, zc, gsum, N, l * HD);
  }

  float* out = (float*)d_out;
  k_l2_zc<<<(N + 7) / 8, 256, 0, stream>>>(zc, out, N);
  k_l2_gc<<<(GN + 7) / 8, 256, 0, stream>>>(gsum, cnt, out + (size_t)N * CD, GN);
}